// OrthMapLayer_15444702397273
// MI455X (gfx1250) — compile-verified
//
#include <hip/hip_runtime.h>
#include <math.h>

typedef __attribute__((ext_vector_type(2))) float v2f;
typedef __attribute__((ext_vector_type(8))) float v8f;

#define D 256
#define Q 64
#define NSUB 20
#define NSWEEP 8

static __device__ __forceinline__ v8f wmma_k4(v2f a, v2f b, v8f c) {
  // D = A(16x4,f32) * B(4x16,f32) + C(16x16,f32)
  return __builtin_amdgcn_wmma_f32_16x16x4_f32(false, a, false, b, (short)0, c,
                                               false, false);
}

__global__ __launch_bounds__(256, 1)
void orthmap_topq_kernel(const float* __restrict__ X, float* __restrict__ out) {
  // Four distinct shared objects (163840 B total) so the compiler has
  // no-alias guarantees between them and can batch/reorder LDS traffic.
  __shared__ float sV[D * Q];  // D x Q  subspace basis
  __shared__ float sW[D * Q];  // D x Q  Y*V (+ small scratch aliases)
  __shared__ float sS[Q * Q];  // Q x Q  Gram / R / Ritz matrix
  __shared__ float sZ[Q * Q];  // Q x Q  Rinv during loop; Jacobi Z after

  const int tid  = (int)threadIdx.x;
  const int lane = tid & 31;
  const int wave = tid >> 5;
  const int b    = (int)blockIdx.x;
  const float* Xb = X + (size_t)b * D * D;

  // Pull this matrix toward L2/WGP$ (reused NSUB+1 times).
  for (int i = tid; i < (D * D) / 32; i += 256)
    __builtin_prefetch(Xb + (size_t)i * 32, 0, 3);

  // ---- Gershgorin-style bound on spectral radius -> shift so Y = X + cI is
  //      PSD-ish and subspace iteration targets the largest-algebraic eigpairs.
  {
    float s = 0.f;
    const float* row = Xb + (size_t)tid * D;
    for (int j = 0; j < D; ++j) s += fabsf(row[j]);
    sW[tid] = s;  // W region is free right now
  }
  __syncthreads();
  if (tid == 0) {
    float m = 0.f;
    for (int i = 0; i < D; ++i) m = fmaxf(m, sW[i]);
    sW[0] = 1.01f * m;
  }
  __syncthreads();
  const float shift = sW[0];
  __syncthreads();

  // ---- deterministic pseudo-random start block V (D x Q)
  for (int j = 0; j < Q; ++j) {
    unsigned h = (unsigned)tid * 1103515245u ^ (unsigned)j * 2654435761u ^ 0x9E3779B9u;
    h ^= h >> 16; h *= 0x7feb352du; h ^= h >> 15; h *= 0x846ca68bu; h ^= h >> 16;
    sV[tid * Q + j] = ((float)(h & 0xFFFFu) * (1.f / 32768.f)) - 1.f;
  }
  __syncthreads();

  // WMMA f32 16x16x4 per-lane fragment coordinates:
  //   A: lanes 0-15 hold M=lane, VGPR0/1 = K=k0,k0+1 with k0 = (lane>>4)*2
  //   B: lanes hold N=lane&15, same K split
  //   C/D: VGPR r holds M = r + 8*(lane>>4), N = lane&15
  const int m15   = lane & 15;
  const int k0    = (lane >> 4) * 2;
  const int rhalf = (lane >> 4) << 3;

  for (int it = 0; it <= NSUB; ++it) {
    // ---------- W = (X + shift*I) * V ---------------------------------------
    // Register-blocked: each wave owns 2 row-tiles (tm); for each tm it keeps
    // 4 accumulators (tn = 0..3) so every X fragment is loaded from global
    // exactly once and feeds 4 back-to-back v_wmma ops.
    for (int tm = wave * 2; tm < wave * 2 + 2; ++tm) {
      const float* arow = Xb + (size_t)(tm * 16 + m15) * D;
      v8f acc[4] = {v8f{}, v8f{}, v8f{}, v8f{}};
      for (int kk = 0; kk < D; kk += 4) {
        v2f a;
        a.x = arow[kk + k0];
        a.y = arow[kk + k0 + 1];
#pragma unroll
        for (int tn = 0; tn < 4; ++tn) {
          const int colg = tn * 16 + m15;
          v2f bb;
          bb.x = sV[(kk + k0) * Q + colg];
          bb.y = sV[(kk + k0 + 1) * Q + colg];
          acc[tn] = wmma_k4(a, bb, acc[tn]);
        }
      }
      // Epilogue: batch all V reads (stride-64 column walk), then FMA+store.
#pragma unroll
      for (int tn = 0; tn < 4; ++tn) {
        const int colg = tn * 16 + m15;
        float vv[8];
#pragma unroll
        for (int r = 0; r < 8; ++r)
          vv[r] = sV[(tm * 16 + rhalf + r) * Q + colg];
#pragma unroll
        for (int r = 0; r < 8; ++r)
          sW[(tm * 16 + rhalf + r) * Q + colg] = acc[tn][r] + shift * vv[r];
      }
    }
    __syncthreads();

    if (it == NSUB) break;  // keep last W = Y*V for Rayleigh-Ritz

    // ---------- G = W^T W  (Q x Q, 4x4 tiles, 2 per wave) -------------------
    for (int t = wave * 2; t < wave * 2 + 2; ++t) {
      const int tm = t >> 2, tn = t & 3;
      const int mg = tm * 16 + m15, colg = tn * 16 + m15;
      v8f acc = {};
      for (int kk = 0; kk < D; kk += 4) {
        v2f a, bb;
        a.x  = sW[(kk + k0) * Q + mg];
        a.y  = sW[(kk + k0 + 1) * Q + mg];
        bb.x = sW[(kk + k0) * Q + colg];
        bb.y = sW[(kk + k0 + 1) * Q + colg];
        acc = wmma_k4(a, bb, acc);
      }
#pragma unroll
      for (int r = 0; r < 8; ++r)
        sS[(tm * 16 + rhalf + r) * Q + colg] = acc[r];
    }
    __syncthreads();

    // ---------- Cholesky G = R^T R (upper R in place) -----------------------
    for (int k = 0; k < Q; ++k) {
      if (tid == 0) sS[k * Q + k] = sqrtf(fmaxf(sS[k * Q + k], 1e-30f));
      __syncthreads();
      const float dk = sS[k * Q + k];
      for (int j = k + 1 + tid; j < Q; j += 256) sS[k * Q + j] /= dk;
      __syncthreads();
      for (int idx = tid; idx < Q * Q; idx += 256) {
        const int i = idx >> 6, j = idx & 63;
        if (i > k && j >= i) sS[i * Q + j] -= sS[k * Q + i] * sS[k * Q + j];
      }
      __syncthreads();
    }

    // ---------- Rinv = R^-1 (upper triangular), into sZ ---------------------
    // One column per thread: thread j reads/writes only column j of sZ
    // (per-lane distinct banks). R entries broadcast-read from sS.
    for (int idx = tid; idx < Q * Q; idx += 256) sZ[idx] = 0.f;
    __syncthreads();
    if (tid < Q) {
      const int j = tid;
      sZ[j * Q + j] = 1.f / sS[j * Q + j];
      for (int i = j - 1; i >= 0; --i) {
        float s = 0.f;
        for (int k = i + 1; k <= j; ++k) s += sS[i * Q + k] * sZ[k * Q + j];
        sZ[i * Q + j] = -s / sS[i * Q + i];
      }
    }
    __syncthreads();

    // ---------- V = W * Rinv  (WMMA GEMM, A reused across 4 tn tiles) -------
    for (int tm = wave * 2; tm < wave * 2 + 2; ++tm) {
      const float* arow = sW + (tm * 16 + m15) * Q;
      v8f acc[4] = {v8f{}, v8f{}, v8f{}, v8f{}};
      for (int kk = 0; kk < Q; kk += 4) {
        v2f a;
        a.x = arow[kk + k0];
        a.y = arow[kk + k0 + 1];
#pragma unroll
        for (int tn = 0; tn < 4; ++tn) {
          const int colg = tn * 16 + m15;
          v2f bb;
          bb.x = sZ[(kk + k0) * Q + colg];
          bb.y = sZ[(kk + k0 + 1) * Q + colg];
          acc[tn] = wmma_k4(a, bb, acc[tn]);
        }
      }
#pragma unroll
      for (int tn = 0; tn < 4; ++tn) {
        const int colg = tn * 16 + m15;
#pragma unroll
        for (int r = 0; r < 8; ++r)
          sV[(tm * 16 + rhalf + r) * Q + colg] = acc[tn][r];
      }
    }
    __syncthreads();
  }

  // ---------- Rayleigh-Ritz: S = V^T W (= V^T Y V) --------------------------
  for (int t = wave * 2; t < wave * 2 + 2; ++t) {
    const int tm = t >> 2, tn = t & 3;
    const int mg = tm * 16 + m15, colg = tn * 16 + m15;
    v8f acc = {};
    for (int kk = 0; kk < D; kk += 4) {
      v2f a, bb;
      a.x  = sV[(kk + k0) * Q + mg];
      a.y  = sV[(kk + k0 + 1) * Q + mg];
      bb.x = sW[(kk + k0) * Q + colg];
      bb.y = sW[(kk + k0 + 1) * Q + colg];
      acc = wmma_k4(a, bb, acc);
    }
#pragma unroll
    for (int r = 0; r < 8; ++r)
      sS[(tm * 16 + rhalf + r) * Q + colg] = acc[r];
  }
  __syncthreads();

  // symmetrize S, init Z = I
  for (int idx = tid; idx < Q * Q; idx += 256) {
    const int i = idx >> 6, j = idx & 63;
    if (i < j) {
      const float m = 0.5f * (sS[i * Q + j] + sS[j * Q + i]);
      sS[i * Q + j] = m;
      sS[j * Q + i] = m;
    }
    sZ[i * Q + j] = (i == j) ? 1.f : 0.f;
  }
  __syncthreads();

  // ---------- cyclic Jacobi on S (64x64), accumulate rotations into Z -------
  // Jacobi scratch aliases the (currently idle) W region.
  float* sRotC  = sW;
  float* sRotS  = sW + 32;
  int*   sPairP = (int*)(sW + 64);
  int*   sPairQ = (int*)(sW + 96);

  for (int sweep = 0; sweep < NSWEEP; ++sweep) {
    for (int rr = 0; rr < Q - 1; ++rr) {  // round-robin: 63 rounds x 32 pairs
      if (tid < 32) {
        int p, q;
        if (tid == 0) { p = Q - 1; q = rr % (Q - 1); }
        else { p = (rr + tid) % (Q - 1); q = (rr + (Q - 1) - tid) % (Q - 1); }
        sPairP[tid] = p;
        sPairQ[tid] = q;
        const float app = sS[p * Q + p], aqq = sS[q * Q + q], apq = sS[p * Q + q];
        float c = 1.f, s = 0.f;
        if (fabsf(apq) > 1e-12f) {
          const float tau = (aqq - app) / (2.f * apq);
          const float tt  = (tau >= 0.f ? 1.f : -1.f) /
                            (fabsf(tau) + sqrtf(1.f + tau * tau));
          c = 1.f / sqrtf(1.f + tt * tt);
          s = tt * c;
        }
        sRotC[tid] = c;
        sRotS[tid] = s;
      }
      __syncthreads();
      const int slot = tid >> 3, sub = tid & 7;
      const int p = sPairP[slot], q = sPairQ[slot];
      const float c = sRotC[slot], s = sRotS[slot];
      // row rotations (disjoint row pairs across slots)
      for (int j = sub * 8; j < sub * 8 + 8; ++j) {
        const float sp = sS[p * Q + j], sq = sS[q * Q + j];
        sS[p * Q + j] = c * sp - s * sq;
        sS[q * Q + j] = s * sp + c * sq;
      }
      __syncthreads();
      // column rotations on S, and accumulate into Z
      for (int i = sub * 8; i < sub * 8 + 8; ++i) {
        const float sp = sS[i * Q + p], sq = sS[i * Q + q];
        sS[i * Q + p] = c * sp - s * sq;
        sS[i * Q + q] = s * sp + c * sq;
        const float zp = sZ[i * Q + p], zq = sZ[i * Q + q];
        sZ[i * Q + p] = c * zp - s * zq;
        sZ[i * Q + q] = s * zp + c * zq;
      }
      __syncthreads();
    }
  }

  // ---------- U = V * Z  (D x Q, WMMA, A reused across 4 tn tiles) ----------
  for (int tm = wave * 2; tm < wave * 2 + 2; ++tm) {
    const float* arow = sV + (tm * 16 + m15) * Q;
    v8f acc[4] = {v8f{}, v8f{}, v8f{}, v8f{}};
    for (int kk = 0; kk < Q; kk += 4) {
      v2f a;
      a.x = arow[kk + k0];
      a.y = arow[kk + k0 + 1];
#pragma unroll
      for (int tn = 0; tn < 4; ++tn) {
        const int colg = tn * 16 + m15;
        v2f bb;
        bb.x = sZ[(kk + k0) * Q + colg];
        bb.y = sZ[(kk + k0 + 1) * Q + colg];
        acc[tn] = wmma_k4(a, bb, acc[tn]);
      }
    }
#pragma unroll
    for (int tn = 0; tn < 4; ++tn) {
      const int colg = tn * 16 + m15;
#pragma unroll
      for (int r = 0; r < 8; ++r)
        sW[(tm * 16 + rhalf + r) * Q + colg] = acc[tn][r];
    }
  }
  __syncthreads();

  // ---------- sort columns by Ritz value (descending) and write out ---------
  if (tid < Q) {
    const float lam = sS[tid * Q + tid];
    int rank = 0;
    for (int k = 0; k < Q; ++k) {
      const float lk = sS[k * Q + k];
      if (lk > lam || (lk == lam && k < tid)) ++rank;
    }
    float* ob = out + (size_t)b * D * Q;
    for (int i = 0; i < D; ++i)
      ob[(size_t)i * Q + rank] = sW[i * Q + tid];
  }
}

extern "C" void kernel_launch(void* const* d_in, const int* in_sizes, int n_in,
                              void* d_out, int out_size, void* d_ws, size_t ws_size,
                              hipStream_t stream) {
  (void)n_in; (void)out_size; (void)d_ws; (void)ws_size;
  const float* X = (const float*)d_in[0];  // (B, 256, 256) f32, symmetric
  // d_in[1] is q (scalar int); reference fixes q = 64, baked into the kernel.
  float* out = (float*)d_out;              // (B, 256, 64) f32
  const int nb = in_sizes[0] / (D * D);    // B = 512
  orthmap_topq_kernel<<<dim3(nb), dim3(256), 0, stream>>>(X, out);
}